// Masker_58153857188550
// MI455X (gfx1250) — compile-verified
//
#include <hip/hip_runtime.h>

// Mask geometry (fixed by the reference module)
#define MASK_H 1024
#define MASK_W 1024
#define TOTAL_BYTES (8u * 1024u * 1024u * 4u)   // B*H*W*sizeof(float) = 32 MB

// Zero-fill tiling: each block clears CHUNK_BYTES via one TDM descriptor that
// iterates an LDS_BYTES tile of zeros TDM_ITERS times.
#define CHUNK_BYTES 16384u
#define LDS_BYTES   4096
#define TDM_ITERS   (CHUNK_BYTES / LDS_BYTES)   // 4 iterations x 4KB
#define FILL_BLOCKS (TOTAL_BYTES / CHUNK_BYTES) // 2048 blocks

typedef float        v2f __attribute__((ext_vector_type(2)));
typedef float        v4f __attribute__((ext_vector_type(4)));
typedef unsigned int v4u __attribute__((ext_vector_type(4)));
typedef int          v4i __attribute__((ext_vector_type(4)));
typedef int          v8i __attribute__((ext_vector_type(8)));

// ---------------------------------------------------------------------------
// Kernel 1: zero the 32MB mask via the Tensor Data Mover. One wave zeroes a
// 4KB LDS tile, then a single TENSOR_STORE_FROM_LDS descriptor (iterate mode,
// lds_addr_increment=0) blasts it 4x across the block's 16KB chunk. Tracked
// with TENSORcnt. Fallback: plain vectorized stores.
// ---------------------------------------------------------------------------
__global__ __launch_bounds__(32) void fill_zero(float* __restrict__ out) {
#if defined(__AMDGCN__) && __has_builtin(__builtin_amdgcn_tensor_store_from_lds) && \
    __has_builtin(__builtin_amdgcn_s_wait_tensorcnt)
  __shared__ unsigned long long zbuf[LDS_BYTES / 8];
  for (int k = threadIdx.x; k < LDS_BYTES / 8; k += 32) zbuf[k] = 0ull;
  // TDM reads LDS directly: make sure the ds_stores have landed.
  asm volatile("s_wait_dscnt 0" ::: "memory");

  unsigned long long gaddr =
      (unsigned long long)out + (unsigned long long)blockIdx.x * CHUNK_BYTES;
  // Generic LDS pointer: low 32 bits are the byte offset in the LDS aperture.
  unsigned lds_off = (unsigned)((unsigned long long)&zbuf[0]);

  // D# group 0: count=1, lds_addr, 57-bit global_addr, type=2 ("image")
  v4u g0 = {1u,
            lds_off,
            (unsigned)(gaddr & 0xffffffffu),
            (unsigned)((gaddr >> 32) & 0x1ffffffu) | (2u << 30)};

  // D# group 1: data_size=3 (8B) | iterate_enable; tensor_dim0=2^30 (no OOB
  // clip), tensor_dim1=1; tile_dim0=512 (8B units = 4KB 1-D tile);
  // tensor_dim0_stride=512.
  v8i g1 = {(3 << 16) | (1 << 19),
            0,
            0x4000 | (1 << 16),
            (int)(512u << 16),
            0,
            512,
            0,
            0};

  // D# group 2 (iterate mode): tensor_dim2=1, lds_addr_increment=0 (reuse the
  // same zero tile), global_addr_increment=512 (8B units = 4KB),
  // iterate_count = TDM_ITERS-1 -> TDM_ITERS iterations.
  v4i g2 = {1, 0, 512, (int)((TDM_ITERS - 1u) << 16)};
  v4i g3 = {0, 0, 0, 0};

  // 6-arg toolchain variant: extra int32x8 group (uncharacterized) -> zeros.
  v8i g4 = {0, 0, 0, 0, 0, 0, 0, 0};

  __builtin_amdgcn_tensor_store_from_lds(g0, g1, g2, g3, g4, 0);
  __builtin_amdgcn_s_wait_tensorcnt(0);
#else
  // Fallback: 32 lanes sweep the block's chunk with 16B stores.
  v4f* dst = (v4f*)(out + (size_t)blockIdx.x * (CHUNK_BYTES / 4u));
  v4f  z   = {0.f, 0.f, 0.f, 0.f};
  const int n4 = CHUNK_BYTES / 16;
  for (int k = threadIdx.x; k < n4; k += 32) dst[k] = z;
#endif
}

// ---------------------------------------------------------------------------
// Kernel 2: one thread per point; write 1.0f into the bounds-clipped 5x5
// patch. Scatter-max of {0,1} into a zeroed mask == plain stores of 1.0 at
// in-bounds patch pixels (write races are benign: same value).
//
// Interior fast path (~99.2% of points): no bounds checks, and each 5-float
// row is written as b32 + 2x b64 (parity-selected so v2f stores are 8B
// aligned) -> 15 memory transactions/point instead of 25. Random points make
// each lane hit its own cacheline, so transaction count is the bottleneck,
// not HBM bandwidth.
// ---------------------------------------------------------------------------
__global__ __launch_bounds__(256) void scatter_mask(const v2f* __restrict__ pts,
                                                    float* __restrict__ mask,
                                                    int N) {
  int i = blockIdx.x * blockDim.x + threadIdx.x;
  if (i >= N) return;
  int b = blockIdx.y;

  // Points are streamed exactly once: nontemporal load keeps L2 for the mask.
  v2f p  = __builtin_nontemporal_load(&pts[(size_t)b * (size_t)N + (size_t)i]);
  int cx = (int)floorf(p.x);
  int cy = (int)floorf(p.y);

  float* img = mask + (size_t)b * (size_t)(MASK_H * MASK_W);

  bool interior = ((unsigned)(cx - 2) <= (unsigned)(MASK_W - 5)) &
                  ((unsigned)(cy - 2) <= (unsigned)(MASK_H - 5));

  if (interior) {
    float* c0 = img + (size_t)(cy - 2) * MASK_W + cx;  // row cy-2, column cx
    const v2f one2 = {1.0f, 1.0f};
    if ((cx & 1) == 0) {
      // cx even: [cx-2,cx-1] and [cx,cx+1] are 8B aligned.
#pragma unroll
      for (int r = 0; r < 5; ++r) {
        float* row = c0 + r * MASK_W;
        *(v2f*)(row - 2) = one2;
        *(v2f*)(row)     = one2;
        row[2]           = 1.0f;
      }
    } else {
      // cx odd: [cx-1,cx] and [cx+1,cx+2] are 8B aligned.
#pragma unroll
      for (int r = 0; r < 5; ++r) {
        float* row = c0 + r * MASK_W;
        row[-2]          = 1.0f;
        *(v2f*)(row - 1) = one2;
        *(v2f*)(row + 1) = one2;
      }
    }
  } else {
    // Boundary: fully bounds-checked scalar stores.
#pragma unroll
    for (int dy = -2; dy <= 2; ++dy) {
      int y = cy + dy;
      if ((unsigned)y < (unsigned)MASK_H) {
        float* row = img + (size_t)y * MASK_W;
#pragma unroll
        for (int dx = -2; dx <= 2; ++dx) {
          int x = cx + dx;
          if ((unsigned)x < (unsigned)MASK_W) row[x] = 1.0f;
        }
      }
    }
  }
}

extern "C" void kernel_launch(void* const* d_in, const int* in_sizes, int n_in,
                              void* d_out, int out_size, void* d_ws, size_t ws_size,
                              hipStream_t stream) {
  const float* pts = (const float*)d_in[0];  // (B, N, 2) fp32
  // d_in[1] = threshold (unused in forward)
  float* out = (float*)d_out;                // (B, H, W) fp32

  int nPts = in_sizes[0] / 2;                // B*N
  int B    = out_size / (MASK_H * MASK_W);   // 8
  int N    = nPts / B;                       // 200000

  fill_zero<<<FILL_BLOCKS, 32, 0, stream>>>(out);

  dim3 grid((N + 255) / 256, B);
  scatter_mask<<<grid, 256, 0, stream>>>((const v2f*)pts, out, N);
}